// TrafficGNN_17188459118980
// MI455X (gfx1250) — compile-verified
//
#include <hip/hip_runtime.h>

// ---------------------------------------------------------------------------
// 2-layer GCN, feature dims 1 -> 16 -> 1, N=200000 nodes, E=12.8M edges.
// Collapsed to scalar edge passes + a WMMA (fp32 16x16x4) per-node MLP.
// ---------------------------------------------------------------------------

typedef float v2f __attribute__((ext_vector_type(2)));
typedef float v8f __attribute__((ext_vector_type(8)));

// K1: deg[i] = 1.0 (self-loop contribution)
__global__ __launch_bounds__(256) void k1_init_deg(float* __restrict__ deg, int N) {
    int i = blockIdx.x * blockDim.x + threadIdx.x;
    if (i < N) deg[i] = 1.0f;
}

// K2: deg[dst] += 1 per edge (float counts exact up to 2^24 > 12.8M)
__global__ __launch_bounds__(256) void k2_degree(const long long* __restrict__ dst,
                                                 float* __restrict__ deg,
                                                 int npairs, long long E) {
    int gid = blockIdx.x * blockDim.x + threadIdx.x;
    int stride = gridDim.x * blockDim.x;
    const longlong2* d2 = reinterpret_cast<const longlong2*>(dst);
    for (int i = gid; i < npairs; i += stride) {
        longlong2 v = d2[i];
        unsafeAtomicAdd(&deg[(int)v.x], 1.0f);
        unsafeAtomicAdd(&deg[(int)v.y], 1.0f);
    }
    if (gid == 0 && (E & 1)) unsafeAtomicAdd(&deg[(int)dst[E - 1]], 1.0f);
}

// K3: dinv = rsqrt(deg); sx = dinv*x; acc1 seeded with self-loop term sx[i]
__global__ __launch_bounds__(256) void k3_norm(const float* __restrict__ x,
                                               const float* __restrict__ deg,
                                               float* __restrict__ dinv,
                                               float* __restrict__ sx,
                                               float* __restrict__ acc1, int N) {
    int i = blockIdx.x * blockDim.x + threadIdx.x;
    if (i < N) {
        float dv = __frsqrt_rn(deg[i]);   // deg >= 1 always (self-loop)
        float s  = dv * x[i];
        dinv[i] = dv;
        sx[i]   = s;
        acc1[i] = s;                      // self-loop: dinv[d]*x[d]; dst scale applied later
    }
}

// K4: acc1[dst] += sx[src]  (dinv[dst] factored out of the segment sum)
__global__ __launch_bounds__(256) void k4_scatter1(const long long* __restrict__ src,
                                                   const long long* __restrict__ dst,
                                                   const float* __restrict__ sx,
                                                   float* __restrict__ acc1,
                                                   int npairs, long long E) {
    int gid = blockIdx.x * blockDim.x + threadIdx.x;
    int stride = gridDim.x * blockDim.x;
    const longlong2* s2 = reinterpret_cast<const longlong2*>(src);
    const longlong2* d2 = reinterpret_cast<const longlong2*>(dst);
    for (int i = gid; i < npairs; i += stride) {
        longlong2 sv = s2[i];
        longlong2 dv = d2[i];
        float m0 = sx[(int)sv.x];
        float m1 = sx[(int)sv.y];
        unsafeAtomicAdd(&acc1[(int)dv.x], m0);
        unsafeAtomicAdd(&acc1[(int)dv.y], m1);
    }
    if (gid == 0 && (E & 1))
        unsafeAtomicAdd(&acc1[(int)dst[E - 1]], sx[(int)src[E - 1]]);
}

// K5: per-node MLP via V_WMMA_F32_16X16X4_F32.
//   agg1[i] = dinv[i]*acc1[i]
//   H[i][k] = relu(W1[k]*agg1[i] + b1[k])        (16 wide)
//   z[i]    = sum_k H[i][k]*W2[k]
// 16 nodes per wave: B[k,n] = H[node_n][k] (nodes on N), A[m,k] = W2[k] (rows equal),
// D[m,n] = z[node_n]; read D VGPR0 on lanes 0..15.
// Layouts (ISA 7.12.2, 32-bit 16x4 tiles): lane l, VGPR v -> K = v + 2*(l>=16),
// A row M = l%16 / B col N = l%16.
__global__ __launch_bounds__(256) void k5_mlp_wmma(const float* __restrict__ acc1,
                                                   const float* __restrict__ dinv,
                                                   const float* __restrict__ W1,
                                                   const float* __restrict__ b1,
                                                   const float* __restrict__ W2,
                                                   float* __restrict__ sz,
                                                   float* __restrict__ out, int N) {
    int tid  = blockIdx.x * blockDim.x + threadIdx.x;
    int wave = tid >> 5;
    int lane = tid & 31;
    int h    = lane >> 4;     // lane-half selects K pair (k = 2h + v)
    int n    = lane & 15;     // column = node within wave's 16-node tile
    int node_base = wave * 16;
    int node  = node_base + n;
    int cnode = node < N ? node : (N - 1);

    float agg = dinv[cnode] * acc1[cnode];

    v8f c = {0.f, 0.f, 0.f, 0.f, 0.f, 0.f, 0.f, 0.f};
#pragma unroll
    for (int cc = 0; cc < 4; ++cc) {       // K chunks of 4: global k = 4*cc + 2*h + v
        int k0 = 4 * cc + 2 * h;
        v2f a, b;
        a[0] = W2[k0];
        a[1] = W2[k0 + 1];
        b[0] = fmaxf(fmaf(W1[k0],     agg, b1[k0]),     0.f);
        b[1] = fmaxf(fmaf(W1[k0 + 1], agg, b1[k0 + 1]), 0.f);
        c = __builtin_amdgcn_wmma_f32_16x16x4_f32(false, a, false, b,
                                                  (short)0, c, false, false);
    }

    if (h == 0 && node < N) {              // D[0][n] lives in VGPR0, lanes 0..15
        float z  = c[0];
        float dv = dinv[node];
        float s  = dv * z;                 // sz = dinv * z
        sz[node]  = s;
        out[node] = s;                     // seed layer-2 accumulator with self-loop term
    }
}

// K6: out[dst] += sz[src]
__global__ __launch_bounds__(256) void k6_scatter2(const long long* __restrict__ src,
                                                   const long long* __restrict__ dst,
                                                   const float* __restrict__ sz,
                                                   float* __restrict__ out,
                                                   int npairs, long long E) {
    int gid = blockIdx.x * blockDim.x + threadIdx.x;
    int stride = gridDim.x * blockDim.x;
    const longlong2* s2 = reinterpret_cast<const longlong2*>(src);
    const longlong2* d2 = reinterpret_cast<const longlong2*>(dst);
    for (int i = gid; i < npairs; i += stride) {
        longlong2 sv = s2[i];
        longlong2 dv = d2[i];
        float m0 = sz[(int)sv.x];
        float m1 = sz[(int)sv.y];
        unsafeAtomicAdd(&out[(int)dv.x], m0);
        unsafeAtomicAdd(&out[(int)dv.y], m1);
    }
    if (gid == 0 && (E & 1))
        unsafeAtomicAdd(&out[(int)dst[E - 1]], sz[(int)src[E - 1]]);
}

// K7: out[i] = b2 + dinv[i] * out[i]   (apply the factored-out dst scale)
__global__ __launch_bounds__(256) void k7_final(const float* __restrict__ dinv,
                                                const float* __restrict__ b2,
                                                float* __restrict__ out, int N) {
    int i = blockIdx.x * blockDim.x + threadIdx.x;
    if (i < N) out[i] = fmaf(dinv[i], out[i], b2[0]);
}

extern "C" void kernel_launch(void* const* d_in, const int* in_sizes, int n_in,
                              void* d_out, int out_size, void* d_ws, size_t ws_size,
                              hipStream_t stream) {
    const float*     x   = (const float*)d_in[0];
    const long long* ei  = (const long long*)d_in[1];   // int64 [2, E]
    const float*     W1  = (const float*)d_in[2];
    const float*     b1  = (const float*)d_in[3];
    const float*     W2  = (const float*)d_in[4];
    const float*     b2  = (const float*)d_in[5];
    float*           out = (float*)d_out;

    const int       N = in_sizes[0];                    // 200000 (x is [N,1])
    const long long E = (long long)(in_sizes[1] / 2);   // 12.8M
    const long long* src = ei;
    const long long* dst = ei + E;
    const int npairs = (int)(E / 2);

    // workspace: 4 x N floats (3.2 MB)
    float* deg  = (float*)d_ws;
    float* dinv = deg  + N;
    float* sx   = dinv + N;   // reused as sz after K4
    float* acc1 = sx   + N;

    const int T = 256;
    const int gN = (N + T - 1) / T;
    const int gE = 4096;                                // grid-stride over edge pairs
    const int waves = (N + 15) / 16;
    const int gW = (waves * 32 + T - 1) / T;

    k1_init_deg<<<gN, T, 0, stream>>>(deg, N);
    k2_degree  <<<gE, T, 0, stream>>>(dst, deg, npairs, E);
    k3_norm    <<<gN, T, 0, stream>>>(x, deg, dinv, sx, acc1, N);
    k4_scatter1<<<gE, T, 0, stream>>>(src, dst, sx, acc1, npairs, E);
    k5_mlp_wmma<<<gW, T, 0, stream>>>(acc1, dinv, W1, b1, W2, /*sz=*/sx, out, N);
    k6_scatter2<<<gE, T, 0, stream>>>(src, dst, /*sz=*/sx, out, npairs, E);
    k7_final   <<<gN, T, 0, stream>>>(dinv, b2, out, N);
}